// fpsLayer_9895604650345
// MI455X (gfx1250) — compile-verified
//
#include <hip/hip_runtime.h>
#include <cstdint>

// Farthest-point sampling for MI455X (gfx1250).
//   B=16 batches, N=65536 points, npoint=1024 picks, out = gathered xyz [B,1024,3] f32.
// One workgroup (1024 threads = 32 wave32 waves) per batch:
//   - dist[N] running-max (squared distances) resident in LDS (256 KB of 320 KB/WGP)
//   - xyz streamed from L2 each step via a 4-buffer / 3-deep GLOBAL_LOAD_ASYNC_TO_LDS
//     pipeline (each thread stages only its own 12 B -> no cross-thread hazards;
//     per-wave s_wait_asynccnt retires exactly the chunk about to be consumed)
//   - argmax with first-index tie-break: wave32 shuffle tournament; the winning point's
//     coordinates ride through the reduction so no global read sits on the serial path.

namespace {

constexpr int kN      = 65536;
constexpr int kNPick  = 1024;
constexpr int kBlock  = 1024;              // 32 waves of 32
constexpr int kChunks = kN / kBlock;       // 64 chunks, 1 point / thread / chunk
constexpr int kNBuf   = 4;                 // async pipeline buffers (3 chunks in flight)

constexpr int kDistBytes = kN * 4;                 // 262144
constexpr int kBufBytes  = kBlock * 12;            // 12288 per stage buffer
constexpr int kSmemBytes = kDistBytes + kNBuf * kBufBytes + 1024;  // ~305 KB < 320 KB

__device__ __forceinline__ void async_ld_b64(uint32_t lds, uint64_t g) {
  asm volatile("global_load_async_to_lds_b64 %0, %1, off"
               :: "v"(lds), "v"(g) : "memory");
}
__device__ __forceinline__ void async_ld_b32(uint32_t lds, uint64_t g) {
  asm volatile("global_load_async_to_lds_b32 %0, %1, off"
               :: "v"(lds), "v"(g) : "memory");
}
// Each chunk = 2 async instructions per wave; waits retire the oldest chunk(s).
__device__ __forceinline__ void wait_async_le6() { asm volatile("s_wait_asynccnt 0x6" ::: "memory"); }
__device__ __forceinline__ void wait_async_le4() { asm volatile("s_wait_asynccnt 0x4" ::: "memory"); }
__device__ __forceinline__ void wait_async_le2() { asm volatile("s_wait_asynccnt 0x2" ::: "memory"); }
__device__ __forceinline__ void wait_async_0()   { asm volatile("s_wait_asynccnt 0x0" ::: "memory"); }

}  // namespace

extern "C" __global__ __launch_bounds__(kBlock, 1)
void fps_kernel(const float* __restrict__ xyz, float* __restrict__ out) {
  extern __shared__ unsigned char smem[];
  float* dist   = reinterpret_cast<float*>(smem);
  float* bufs   = reinterpret_cast<float*>(smem + kDistBytes);          // kNBuf contiguous buffers
  float* redv   = reinterpret_cast<float*>(smem + kDistBytes + kNBuf * kBufBytes);
  int*   redi   = reinterpret_cast<int*>(redv + 32);
  float* redx   = reinterpret_cast<float*>(redi + 32);
  float* redy   = redx + 32;
  float* redz   = redy + 32;
  float* bcast  = redz + 32;

  const int tid  = threadIdx.x;
  const int lane = tid & 31;
  const int wv   = tid >> 5;
  const int b    = blockIdx.x;

  const float*   xb     = xyz + (size_t)b * kN * 3;
  const uint64_t gbase  = (uint64_t)(uintptr_t)xb + (uint64_t)tid * 12u;
  const uint32_t lbase  = (uint32_t)(uintptr_t)bufs + (uint32_t)tid * 12u;  // LDS byte addr

  // ---- init: dist = 0, first pick is index 0 (reference semantics) ----
#pragma unroll 4
  for (int c = 0; c < kChunks; ++c) dist[c * kBlock + tid] = 0.0f;
  if (tid == 0) {
    float x = xb[0], y = xb[1], z = xb[2];
    bcast[0] = x; bcast[1] = y; bcast[2] = z;
    float* o = out + (size_t)b * kNPick * 3;
    o[0] = x; o[1] = y; o[2] = z;
  }
  __syncthreads();
  float lx = bcast[0], ly = bcast[1], lz = bcast[2];

  // ---- 1023 serial FPS steps ----
  for (int j = 1; j < kNPick; ++j) {
    float bestv = -1.0f;
    int   besti = 0x7fffffff;
    float bx = 0.f, by = 0.f, bz = 0.f;

    auto stage = [&](int c) {
      uint64_t g = gbase + (uint64_t)(c * kBlock) * 12u;
      uint32_t l = lbase + (uint32_t)(c & (kNBuf - 1)) * (uint32_t)kBufBytes;
      async_ld_b64(l, g);
      async_ld_b32(l + 8u, g + 8u);
    };
    auto process = [&](int c) {
      const float* q = bufs + (size_t)(c & (kNBuf - 1)) * (kBlock * 3) + tid * 3;
      float qx = q[0], qy = q[1], qz = q[2];
      float dx = qx - lx, dy = qy - ly, dz = qz - lz;
      float d  = fmaf(dx, dx, fmaf(dy, dy, dz * dz));  // squared dist (sqrt monotone)
      int   p  = c * kBlock + tid;
      float nd = fmaxf(dist[p], d);                    // running max (per reference)
      dist[p] = nd;
      if (nd > bestv || (nd == bestv && p < besti)) {
        bestv = nd; besti = p; bx = qx; by = qy; bz = qz;
      }
    };

    // prologue: put 3 chunks in flight
    stage(0); stage(1); stage(2);
    for (int c = 0; c < kChunks - 3; ++c) {
      stage(c + 3);
      wait_async_le6();   // retire chunk c (3 chunks remain in flight)
      process(c);
    }
    wait_async_le4(); process(kChunks - 3);
    wait_async_le2(); process(kChunks - 2);
    wait_async_0();   process(kChunks - 1);

    // wave32 argmax tournament; coordinates ride along (no global re-read)
#pragma unroll
    for (int off = 16; off > 0; off >>= 1) {
      float ov = __shfl_xor(bestv, off, 32);
      int   oi = __shfl_xor(besti, off, 32);
      float ox = __shfl_xor(bx, off, 32);
      float oy = __shfl_xor(by, off, 32);
      float oz = __shfl_xor(bz, off, 32);
      if (ov > bestv || (ov == bestv && oi < besti)) {
        bestv = ov; besti = oi; bx = ox; by = oy; bz = oz;
      }
    }
    if (lane == 0) {
      redv[wv] = bestv; redi[wv] = besti;
      redx[wv] = bx; redy[wv] = by; redz[wv] = bz;
    }
    __syncthreads();
    if (wv == 0) {  // 32 wave results -> one wave32
      bestv = redv[lane]; besti = redi[lane];
      bx = redx[lane]; by = redy[lane]; bz = redz[lane];
#pragma unroll
      for (int off = 16; off > 0; off >>= 1) {
        float ov = __shfl_xor(bestv, off, 32);
        int   oi = __shfl_xor(besti, off, 32);
        float ox = __shfl_xor(bx, off, 32);
        float oy = __shfl_xor(by, off, 32);
        float oz = __shfl_xor(bz, off, 32);
        if (ov > bestv || (ov == bestv && oi < besti)) {
          bestv = ov; besti = oi; bx = ox; by = oy; bz = oz;
        }
      }
      if (lane == 0) {
        bcast[0] = bx; bcast[1] = by; bcast[2] = bz;
        float* o = out + ((size_t)b * kNPick + j) * 3;
        o[0] = bx; o[1] = by; o[2] = bz;
      }
    }
    __syncthreads();
    lx = bcast[0]; ly = bcast[1]; lz = bcast[2];
  }
}

extern "C" void kernel_launch(void* const* d_in, const int* in_sizes, int n_in,
                              void* d_out, int out_size, void* d_ws, size_t ws_size,
                              hipStream_t stream) {
  const float* xyz = (const float*)d_in[0];
  float* out = (float*)d_out;
  int batches = in_sizes[0] / (kN * 3);   // = 16 for the reference shapes
  if (batches <= 0) batches = 1;
  fps_kernel<<<dim3(batches), dim3(kBlock), kSmemBytes, stream>>>(xyz, out);
  (void)n_in; (void)out_size; (void)d_ws; (void)ws_size;
}